// Lingunet5Filter_49469433315712
// MI455X (gfx1250) — compile-verified
//
#include <hip/hip_runtime.h>

typedef __attribute__((ext_vector_type(2))) float v2f;
typedef __attribute__((ext_vector_type(8))) float v8f;
typedef __attribute__((ext_vector_type(4))) int   v4i;

typedef __attribute__((address_space(1))) v4i* gv4i_p;   // global int4*
typedef __attribute__((address_space(3))) v4i* sv4i_p;   // LDS int4*

#define NB 32     // batch
#define NC 128    // channels (Cin == Cout)

#define MODE_RAW   0   // store Y unnormalized (level 5)
#define MODE_SUMS  1   // no store; accumulate per-channel sum/sumsq
#define MODE_NORM  2   // recompute; store (y-mean)*rstd using precomputed sums

// ---------------------------------------------------------------------------
// Zero the per-(b,channel) sum / sumsq accumulators (2*NB*NC floats = 32 KB).
// ---------------------------------------------------------------------------
__global__ __launch_bounds__(256) void zero_ws_kernel(float* __restrict__ ws) {
  int i = blockIdx.x * 256 + threadIdx.x;
  if (i < 2 * NB * NC) ws[i] = 0.0f;
}

// ---------------------------------------------------------------------------
// Batched dynamic 1x1 conv:  Y[b] = F[b] (128x128) * X[b] (128xS)
// Grid: (S/64, NB), block: 256 threads = 8 wave32.
// Wave w computes output channels [16w,16w+16) x 64 spatial columns with
// v_wmma_f32_16x16x4_f32, K=128 accumulated in steps of 4.
// ---------------------------------------------------------------------------
template <int MODE>
__global__ __launch_bounds__(256) void dynconv_wmma_kernel(
    const float* __restrict__ X, const float* __restrict__ F,
    float* __restrict__ Y, float* __restrict__ sums, int S, float invS)
{
  __shared__ float Fs[NC * NC];          // 64 KB: full F[b], row-major [o][c]

  const int b    = blockIdx.y;
  const int n0   = blockIdx.x * 64;      // spatial base column
  const int wave = threadIdx.x >> 5;
  const int lane = threadIdx.x & 31;
  const int half = lane >> 4;            // 0: lanes 0-15, 1: lanes 16-31
  const int l16  = lane & 15;

  // ---- Stage F[b] (16384 floats) into LDS -------------------------------
  const float* Fg = F + (size_t)b * NC * NC;
#if __has_builtin(__builtin_amdgcn_global_load_async_to_lds_b128) && \
    __has_builtin(__builtin_amdgcn_s_wait_asynccnt)
  // gfx1250 async copy: global -> LDS without a VGPR round trip.
  for (int i = threadIdx.x * 4; i < NC * NC; i += 256 * 4) {
    __builtin_amdgcn_global_load_async_to_lds_b128(
        (gv4i_p)(uintptr_t)(Fg + i),
        (sv4i_p)&Fs[i], 0, 0);
  }
  __builtin_amdgcn_s_wait_asynccnt(0);
#else
  for (int i = threadIdx.x * 4; i < NC * NC; i += 256 * 4)
    *(float4*)&Fs[i] = *(const float4*)&Fg[i];
#endif
  __syncthreads();

  const int o_base = wave * 16;
  const float* Xb = X + (size_t)b * NC * S + n0 + l16;

  v8f acc0 = {}, acc1 = {}, acc2 = {}, acc3 = {};

  #pragma unroll 4
  for (int k = 0; k < NC; k += 4) {
    const int kk = k + 2 * half;
    // A fragment (16x4 of F): lane holds F[o_base+l16][kk .. kk+1]
    v2f a = *(const v2f*)&Fs[(o_base + l16) * NC + kk];
    // B fragments (4x16 of X): lane holds X[kk][n], X[kk+1][n]
    const float* xr = Xb + (size_t)kk * S;
    v2f b0; b0.x = xr[0];  b0.y = xr[0 + S];
    v2f b1; b1.x = xr[16]; b1.y = xr[16 + S];
    v2f b2; b2.x = xr[32]; b2.y = xr[32 + S];
    v2f b3; b3.x = xr[48]; b3.y = xr[48 + S];
    acc0 = __builtin_amdgcn_wmma_f32_16x16x4_f32(false, a, false, b0, (short)0, acc0, false, false);
    acc1 = __builtin_amdgcn_wmma_f32_16x16x4_f32(false, a, false, b1, (short)0, acc1, false, false);
    acc2 = __builtin_amdgcn_wmma_f32_16x16x4_f32(false, a, false, b2, (short)0, acc2, false, false);
    acc3 = __builtin_amdgcn_wmma_f32_16x16x4_f32(false, a, false, b3, (short)0, acc3, false, false);
  }

  // D layout: acc_t[v] = Y[o_base + v + 8*half][n0 + 16*t + l16]
  if (MODE == MODE_SUMS) {
    float rs[8], rq[8];
    #pragma unroll
    for (int v = 0; v < 8; ++v) { rs[v] = 0.0f; rq[v] = 0.0f; }
    auto gather = [&](const v8f& acc) {
      #pragma unroll
      for (int v = 0; v < 8; ++v) {
        float val = acc[v];
        rs[v] += val;
        rq[v] += val * val;
      }
    };
    gather(acc0); gather(acc1); gather(acc2); gather(acc3);
    #pragma unroll
    for (int v = 0; v < 8; ++v) {
      float s = rs[v], q = rq[v];
      // Reduce across the 16 lanes sharing this output channel.
      #pragma unroll
      for (int m = 8; m >= 1; m >>= 1) {
        s += __shfl_xor(s, m, 32);
        q += __shfl_xor(q, m, 32);
      }
      if (l16 == 0) {
        int ch = b * NC + o_base + v + 8 * half;
        atomicAdd(&sums[ch], s);
        atomicAdd(&sums[NB * NC + ch], q);
      }
    }
  } else {
    float mean[8], rstd[8];
    if (MODE == MODE_NORM) {
      #pragma unroll
      for (int v = 0; v < 8; ++v) {
        int ch = b * NC + o_base + v + 8 * half;
        float m = sums[ch] * invS;
        float q = sums[NB * NC + ch] * invS;
        mean[v] = m;
        rstd[v] = rsqrtf(fmaxf(q - m * m, 0.0f) + 1e-5f);
      }
    }
    float* Yb = Y + ((size_t)b * NC + o_base + 8 * half) * S + n0 + l16;
    auto emit = [&](const v8f& acc, int t) {
      #pragma unroll
      for (int v = 0; v < 8; ++v) {
        float val = acc[v];
        if (MODE == MODE_NORM) val = (val - mean[v]) * rstd[v];
        Yb[(size_t)v * S + t * 16] = val;
      }
    };
    emit(acc0, 0); emit(acc1, 1); emit(acc2, 2); emit(acc3, 3);
  }
}

// ---------------------------------------------------------------------------
extern "C" void kernel_launch(void* const* d_in, const int* in_sizes, int n_in,
                              void* d_out, int out_size, void* d_ws, size_t ws_size,
                              hipStream_t stream) {
  (void)in_sizes; (void)n_in; (void)out_size; (void)ws_size;
  static const int sizes[5] = {128 * 128, 64 * 64, 32 * 32, 16 * 16, 8 * 8};

  float* out = (float*)d_out;
  float* ws  = (float*)d_ws;
  size_t off = 0;

  for (int lvl = 0; lvl < 5; ++lvl) {
    const float* X = (const float*)d_in[2 * lvl];       // x{lvl+1}
    const float* F = (const float*)d_in[2 * lvl + 1];   // l{lvl+1}fs
    float* Y = out + off;
    const int S = sizes[lvl];
    const float invS = 1.0f / (float)S;
    dim3 grid(S / 64, NB);

    if (lvl < 4) {
      // Pass 0: zero stats accumulators.
      zero_ws_kernel<<<(2 * NB * NC + 255) / 256, 256, 0, stream>>>(ws);
      // Pass A: compute per-(b,channel) sum / sumsq (no Y traffic).
      dynconv_wmma_kernel<MODE_SUMS><<<grid, 256, 0, stream>>>(X, F, Y, ws, S, invS);
      // Pass B: recompute GEMM, normalize at store time — Y written once.
      dynconv_wmma_kernel<MODE_NORM><<<grid, 256, 0, stream>>>(X, F, Y, ws, S, invS);
    } else {
      // Level 5: no instance norm.
      dynconv_wmma_kernel<MODE_RAW><<<grid, 256, 0, stream>>>(X, F, Y, ws, S, invS);
    }

    off += (size_t)NB * NC * S;
  }
}